// ByteSequenceEmbedder_89249420411016
// MI455X (gfx1250) — compile-verified
//
#include <hip/hip_runtime.h>
#include <hip/hip_bf16.h>
#include <math.h>

// ---------------------------------------------------------------------------
// Problem constants
// ---------------------------------------------------------------------------
#define BSZ    16
#define NW     1024
#define T_SEQ  3072
#define BED    128
#define WED    512
#define M_ROWS (BSZ * T_SEQ)   // 49152 token rows
#define P_ROWS (BSZ * NW)      // 16384 pooled rows

// LDS staging geometry
#define KT   16     // K tile depth
#define SD   20     // padded col stride (dwords) for [col][k] tiles: conflict-free b64 reads
#define KNS  144    // padded row stride (dwords) for [k][n] conv tiles: +2 rows => +32 banks
#define BUFD 2560   // per-buffer LDS dwords (max of both layouts); 2 buffers = 20 KB

typedef __attribute__((ext_vector_type(2))) float v2f;
typedef __attribute__((ext_vector_type(8))) float v8f;

// GCC-style 4xi32 vector: exact pointee type expected by the async-LDS builtin
typedef int v4i_gcc __attribute__((vector_size(16)));
typedef __attribute__((address_space(1))) v4i_gcc* g128_t;   // global (as1)
typedef __attribute__((address_space(3))) v4i_gcc* l128_t;   // LDS (as3)

// ---------------------------------------------------------------------------
// CDNA5 async global->LDS copy (ASYNCcnt) with graceful fallback
// ---------------------------------------------------------------------------
#if defined(__has_builtin)
#  if __has_builtin(__builtin_amdgcn_global_load_async_to_lds_b128)
#    define HAVE_ASYNC_LDS 1
#  endif
#  if __has_builtin(__builtin_amdgcn_s_wait_asynccnt)
#    define HAVE_WAIT_ASYNC 1
#  endif
#endif
#ifndef HAVE_ASYNC_LDS
#  define HAVE_ASYNC_LDS 0
#endif
#ifndef HAVE_WAIT_ASYNC
#  define HAVE_WAIT_ASYNC 0
#endif

__device__ __forceinline__ void async_cp16(const float* g, float* l) {
#if HAVE_ASYNC_LDS
  __builtin_amdgcn_global_load_async_to_lds_b128((g128_t)(g), (l128_t)(l), 0, 0);
#else
  *(float4*)l = *(const float4*)g;   // sync staging fallback (vmem + ds_store)
#endif
}

__device__ __forceinline__ void wait_async_2() {
#if HAVE_ASYNC_LDS
#  if HAVE_WAIT_ASYNC
  __builtin_amdgcn_s_wait_asynccnt(2);
#  else
  asm volatile("s_wait_asynccnt 2" ::: "memory");
#  endif
#endif
}
__device__ __forceinline__ void wait_async_0() {
#if HAVE_ASYNC_LDS
#  if HAVE_WAIT_ASYNC
  __builtin_amdgcn_s_wait_asynccnt(0);
#  else
  asm volatile("s_wait_asynccnt 0" ::: "memory");
#  endif
#endif
}

// V_WMMA_F32_16X16X4_F32 (codegen-confirmed in round 1)
__device__ __forceinline__ v8f wmma4(v2f a, v2f b, v8f c) {
  return __builtin_amdgcn_wmma_f32_16x16x4_f32(false, a, false, b, (short)0, c,
                                               false, false);
}
__device__ __forceinline__ v8f vzero8() {
  v8f z = {0.f, 0.f, 0.f, 0.f, 0.f, 0.f, 0.f, 0.f};
  return z;
}

// ---------------------------------------------------------------------------
// 1) Embedding gather
// ---------------------------------------------------------------------------
__global__ __launch_bounds__(256) void embed_kernel(
    const int* __restrict__ tok, const int* __restrict__ mask,
    const float* __restrict__ emb, float* __restrict__ x0) {
  long i = (long)blockIdx.x * 256 + threadIdx.x;   // over M_ROWS*BED
  int  c = (int)(i & (BED - 1));
  long r = i >> 7;                                  // BED == 128
  int  t  = tok[r];
  int  mk = mask[r] ? 4 : 0;                        // BPE_MASK_IDX = 4
  x0[i] = emb[(long)t * BED + c] + emb[(long)mk * BED + c];
}

// ---------------------------------------------------------------------------
// 2) Per-batch cumsum of pool_lengths
// ---------------------------------------------------------------------------
__global__ void cumsum_kernel(const int* __restrict__ pl, int* __restrict__ cum) {
  int b = threadIdx.x;
  if (b >= BSZ) return;
  int acc = 0;
  for (int s = 0; s < NW; ++s) { acc += pl[b * NW + s]; cum[b * NW + s] = acc; }
}

// ---------------------------------------------------------------------------
// 3) fp32 WMMA GEMM with double-buffered async-LDS weight tiles.
//    A: Mtotal x K activations (direct from global; rows reused across K loop).
//    B: TAPS x K x N ("KN" conv weights) or N x K ("NK", y = x @ W.T).
//    Block 256thr = 8 waves (4M x 2N); wave tile 32M x 64N; block 128 x 128.
// ---------------------------------------------------------------------------
template <int TAPS, bool B_NK, bool RELU, bool RESID>
__global__ __launch_bounds__(256) void gemm_wmma_kernel(
    const float* __restrict__ A, const float* __restrict__ B,
    const float* __restrict__ bias, const float* __restrict__ resid,
    float* __restrict__ C, int Mtotal, int K, int N) {
  __shared__ float smem[2 * BUFD];
  const int tid   = threadIdx.x;
  const int lane  = tid & 31;
  const int wave  = tid >> 5;
  const int waveM = wave >> 1;                 // 0..3
  const int waveN = wave & 1;                  // 0..1
  const int m0    = blockIdx.x * 128 + waveM * 32;
  const int nb0   = blockIdx.y * 128;
  const int lhalf = lane >> 4;
  const int lmod  = lane & 15;
  const int koff  = lhalf * 2;

  const int NIT = TAPS * (K / KT);             // total K tiles across taps

  // Cooperative weight-tile fetch: 512 x 16B chunks, 2 per thread.
  auto issueB = [&](int tp, int k0, int bufSel) {
    float* buf = smem + bufSel * BUFD;
#pragma unroll
    for (int rep = 0; rep < 2; ++rep) {
      const int q = tid + rep * 256;           // 0..511
      if (B_NK) {
        const int col = q >> 2, ch = (q & 3) * 4;
        async_cp16(B + (long)(nb0 + col) * K + k0 + ch, buf + col * SD + ch);
      } else {
        const int row = q >> 5, ch = (q & 31) * 4;
        async_cp16(B + (long)tp * K * N + (long)(k0 + row) * N + nb0 + ch,
                   buf + row * KNS + ch);
      }
    }
  };

  v8f acc[2][4];
#pragma unroll
  for (int i = 0; i < 2; ++i)
#pragma unroll
    for (int j = 0; j < 4; ++j) acc[i][j] = vzero8();

  const int r0 = m0 + lmod;                    // A rows (lane = M)
  const int r1 = r0 + 16;

  int tap = 0, k0 = 0;
  issueB(0, 0, 0);

  for (int it = 0; it < NIT; ++it) {
    int ntap = tap, nk0 = k0 + KT;             // next tile coords
    if (nk0 == K) { nk0 = 0; ++ntap; }

    // per-tap A row masking (SAME padding within each batch row), branchless
    const int shift = (TAPS == 3) ? (tap - 1) : 0;
    bool ok0 = true, ok1 = true;
    long ar0 = r0, ar1 = r1;
    if (TAPS == 3) {
      const int t0 = r0 % T_SEQ + shift;
      const int t1 = r1 % T_SEQ + shift;
      ok0 = (t0 >= 0) & (t0 < T_SEQ);
      ok1 = (t1 >= 0) & (t1 < T_SEQ);
      ar0 = r0 + shift; ar1 = r1 + shift;
      if (ar0 < 0) ar0 = 0; if (ar0 > Mtotal - 1) ar0 = Mtotal - 1;
      if (ar1 < 0) ar1 = 0; if (ar1 > Mtotal - 1) ar1 = Mtotal - 1;
    }
    const float* a0p = A + ar0 * (long)K + k0;
    const float* a1p = A + ar1 * (long)K + k0;
    const float* sb  = smem + (it & 1) * BUFD;

    if (it + 1 < NIT) { issueB(ntap, nk0, (it + 1) & 1); wait_async_2(); }
    else              { wait_async_0(); }
    __syncthreads();                           // publish tile `it` to all waves

#pragma unroll
    for (int kk = 0; kk < KT; kk += 4) {
      const int kc = kk + koff;
      v2f a0 = *(const v2f*)(a0p + kc);
      v2f a1 = *(const v2f*)(a1p + kc);
      if (TAPS == 3) {
        a0.x = ok0 ? a0.x : 0.f; a0.y = ok0 ? a0.y : 0.f;
        a1.x = ok1 ? a1.x : 0.f; a1.y = ok1 ? a1.y : 0.f;
      }
#pragma unroll
      for (int j = 0; j < 4; ++j) {
        const int nl = waveN * 64 + j * 16 + lmod;   // col within block tile
        v2f bf;
        if (B_NK) {
          bf = *(const v2f*)(sb + nl * SD + kc);     // ds_load_b64, pad=20
        } else {
          bf.x = sb[kc * KNS + nl];                  // ds_load_b32 x2, pad=144
          bf.y = sb[(kc + 1) * KNS + nl];
        }
        acc[0][j] = wmma4(a0, bf, acc[0][j]);
        acc[1][j] = wmma4(a1, bf, acc[1][j]);
      }
    }
    __syncthreads();                           // tile read done; buffer reusable
    tap = ntap; k0 = nk0;
  }

  // C layout: lane = N; VGPR r -> M = r (lanes 0-15) / r+8 (lanes 16-31)
  const int rbase = m0 + lhalf * 8;
#pragma unroll
  for (int i = 0; i < 2; ++i)
#pragma unroll
    for (int j = 0; j < 4; ++j) {
      const int n  = nb0 + waveN * 64 + j * 16 + lmod;
      const float bv = bias[n];
#pragma unroll
      for (int r = 0; r < 8; ++r) {
        const int row = rbase + i * 16 + r;
        float v = acc[i][j][r] + bv;
        if (RELU)  v = fmaxf(v, 0.f);
        if (RESID) v += resid[(long)row * N + n];
        C[(long)row * N + n] = v;
      }
    }
}

// ---------------------------------------------------------------------------
// 4) Fused highway layer (dual-B WMMA + gate epilogue), async-LDS W tiles.
//    x_out = sigmoid(g)*relu(h) + (1-sigmoid(g))*x ; W:(2*WED, WED) row-major.
//    Block 128M x 64N, wave tile 32M x 32N (h and g accumulated together).
// ---------------------------------------------------------------------------
__global__ __launch_bounds__(256) void highway_wmma_kernel(
    const float* __restrict__ Xin, const float* __restrict__ W,
    const float* __restrict__ bvec, float* __restrict__ Xout) {
  __shared__ float smem[2 * BUFD];             // 2 bufs x (2 mats x 64 cols) x SD
  const int tid   = threadIdx.x;
  const int lane  = tid & 31;
  const int wave  = tid >> 5;
  const int waveM = wave >> 1;
  const int waveN = wave & 1;
  const int m0    = blockIdx.x * 128 + waveM * 32;
  const int nb0   = blockIdx.y * 64;
  const int lhalf = lane >> 4;
  const int lmod  = lane & 15;
  const int koff  = lhalf * 2;

  auto issueW = [&](int k0, int bufSel) {
    float* buf = smem + bufSel * BUFD;
#pragma unroll
    for (int rep = 0; rep < 2; ++rep) {
      const int q   = tid + rep * 256;         // 0..511
      const int mat = q >> 8;                  // 0=h rows, 1=g rows (+WED)
      const int rem = q & 255;
      const int col = rem >> 2, ch = (rem & 3) * 4;
      async_cp16(W + (long)(mat * WED + nb0 + col) * WED + k0 + ch,
                 buf + (mat * 64 + col) * SD + ch);
    }
  };

  v8f ah[2][2], ag[2][2];
#pragma unroll
  for (int i = 0; i < 2; ++i)
#pragma unroll
    for (int j = 0; j < 2; ++j) { ah[i][j] = vzero8(); ag[i][j] = vzero8(); }

  const float* a0p = Xin + (long)(m0 + lmod) * WED;
  const float* a1p = a0p + 16L * WED;

  const int NIT = WED / KT;                    // 32 tiles
  issueW(0, 0);
  for (int it = 0; it < NIT; ++it) {
    const int k0 = it * KT;
    const float* sb = smem + (it & 1) * BUFD;

    if (it + 1 < NIT) { issueW(k0 + KT, (it + 1) & 1); wait_async_2(); }
    else              { wait_async_0(); }
    __syncthreads();

#pragma unroll
    for (int kk = 0; kk < KT; kk += 4) {
      const int kc = kk + koff;
      v2f a0 = *(const v2f*)(a0p + k0 + kc);
      v2f a1 = *(const v2f*)(a1p + k0 + kc);
#pragma unroll
      for (int j = 0; j < 2; ++j) {
        const int cl = waveN * 32 + j * 16 + lmod;       // col within 64-tile
        v2f bh = *(const v2f*)(sb + cl * SD + kc);
        v2f bg = *(const v2f*)(sb + (64 + cl) * SD + kc);
        ah[0][j] = wmma4(a0, bh, ah[0][j]);
        ah[1][j] = wmma4(a1, bh, ah[1][j]);
        ag[0][j] = wmma4(a0, bg, ag[0][j]);
        ag[1][j] = wmma4(a1, bg, ag[1][j]);
      }
    }
    __syncthreads();
  }

  const int rbase = m0 + lhalf * 8;
#pragma unroll
  for (int i = 0; i < 2; ++i)
#pragma unroll
    for (int j = 0; j < 2; ++j) {
      const int n  = nb0 + waveN * 32 + j * 16 + lmod;
      const float bh = bvec[n];
      const float bg = bvec[WED + n];
#pragma unroll
      for (int r = 0; r < 8; ++r) {
        const int row = rbase + i * 16 + r;
        const float h  = ah[i][j][r] + bh;
        const float gz = ag[i][j][r] + bg;
        const float g  = 1.f / (1.f + __expf(-gz));
        const float xo = Xin[(long)row * WED + n];
        Xout[(long)row * WED + n] = g * fmaxf(h, 0.f) + (1.f - g) * xo;
      }
    }
}

// ---------------------------------------------------------------------------
// 5) Segment-max pooling
// ---------------------------------------------------------------------------
__global__ __launch_bounds__(128) void pool_kernel(
    const float* __restrict__ x, const int* __restrict__ cum,
    float* __restrict__ pooled) {
  const int g = blockIdx.x;            // 0 .. P_ROWS-1
  const int b = g >> 10;               // NW == 1024
  const int s = g & (NW - 1);
  const int end   = cum[b * NW + s];
  const int start = (s == 0) ? 0 : cum[b * NW + s - 1];
  const int c0 = threadIdx.x * 4;
  float a0 = -__builtin_huge_valf(), a1 = a0, a2 = a0, a3 = a0;
  for (int t = start; t < end; ++t) {
    const float* row = x + ((long)b * T_SEQ + t) * WED + c0;
    a0 = fmaxf(a0, row[0]); a1 = fmaxf(a1, row[1]);
    a2 = fmaxf(a2, row[2]); a3 = fmaxf(a3, row[3]);
  }
  if (end <= start) { a0 = a1 = a2 = a3 = 0.f; }   // isfinite clean-up
  float* o = pooled + (long)g * WED + c0;
  o[0] = a0; o[1] = a1; o[2] = a2; o[3] = a3;
}

// ---------------------------------------------------------------------------
// Orchestration (ws: buf0 | buf1 | cum). Dataflow:
// emb->buf1 | conv0: buf1->buf0 | hw0: buf0->buf1->buf0 | conv1(+res): buf0->buf1
// | hw1: buf1->buf0->buf1 | pool: buf1->buf0 | proj: buf0->d_out
// ---------------------------------------------------------------------------
extern "C" void kernel_launch(void* const* d_in, const int* in_sizes, int n_in,
                              void* d_out, int out_size, void* d_ws, size_t ws_size,
                              hipStream_t stream) {
  const int*   byte_tokens = (const int*)d_in[0];
  const int*   bpe_mask    = (const int*)d_in[1];
  const int*   pool_len    = (const int*)d_in[2];
  const float* tok_emb     = (const float*)d_in[3];
  const float* conv0_W     = (const float*)d_in[4];
  const float* conv0_b     = (const float*)d_in[5];
  const float* conv1_W     = (const float*)d_in[6];
  const float* conv1_b     = (const float*)d_in[7];
  const float* hw0_W       = (const float*)d_in[8];
  const float* hw0_b       = (const float*)d_in[9];
  const float* hw1_W       = (const float*)d_in[10];
  const float* hw1_b       = (const float*)d_in[11];
  const float* proj_W      = (const float*)d_in[12];
  const float* proj_b      = (const float*)d_in[13];
  float* out = (float*)d_out;

  float* buf0 = (float*)d_ws;
  float* buf1 = buf0 + (size_t)M_ROWS * WED;
  int*   cum  = (int*)(buf1 + (size_t)M_ROWS * WED);
  float* x0   = buf1;   // embeddings (M_ROWS x BED) parked in buf1

  {
    const long total = (long)M_ROWS * BED;
    embed_kernel<<<dim3((unsigned)(total / 256)), dim3(256), 0, stream>>>(
        byte_tokens, bpe_mask, tok_emb, x0);
  }
  cumsum_kernel<<<dim3(1), dim3(32), 0, stream>>>(pool_len, cum);

  gemm_wmma_kernel<3, false, true, false>
      <<<dim3(M_ROWS / 128, WED / 128), dim3(256), 0, stream>>>(
          x0, conv0_W, conv0_b, nullptr, buf0, M_ROWS, BED, WED);

  const dim3 hgrid(M_ROWS / 128, WED / 64);
  highway_wmma_kernel<<<hgrid, dim3(256), 0, stream>>>(buf0, hw0_W, hw0_b, buf1);
  highway_wmma_kernel<<<hgrid, dim3(256), 0, stream>>>(
      buf1, hw0_W + (size_t)2 * WED * WED, hw0_b + 2 * WED, buf0);

  gemm_wmma_kernel<3, false, true, true>
      <<<dim3(M_ROWS / 128, WED / 128), dim3(256), 0, stream>>>(
          buf0, conv1_W, conv1_b, buf0, buf1, M_ROWS, WED, WED);

  highway_wmma_kernel<<<hgrid, dim3(256), 0, stream>>>(buf1, hw1_W, hw1_b, buf0);
  highway_wmma_kernel<<<hgrid, dim3(256), 0, stream>>>(
      buf0, hw1_W + (size_t)2 * WED * WED, hw1_b + 2 * WED, buf1);

  pool_kernel<<<dim3(P_ROWS), dim3(128), 0, stream>>>(buf1, cum, buf0);

  gemm_wmma_kernel<1, true, false, false>
      <<<dim3(P_ROWS / 128, WED / 128), dim3(256), 0, stream>>>(
          buf0, proj_W, proj_b, nullptr, out, P_ROWS, WED, WED);
}